// JointRCNN_14465449853476
// MI455X (gfx1250) — compile-verified
//
#include <hip/hip_runtime.h>
#include <cstdint>
#include <cstddef>

// -----------------------------------------------------------------------------
// JointRCNN postprocess for MI455X (gfx1250).
// Memory-bound (~0.4 flop/byte, ~1.18 GB traffic -> ~50us floor at 23.3 TB/s).
// wmma intentionally 0: there is no matmul content in this op. CDNA5 paths:
//   * wave32 cross-lane softmax reduction (one row per wave, 3 classes/lane)
//   * global_load_async_to_lds_b128 / global_store_async_from_lds_b128
//     + s_wait_asynccnt (ASYNCcnt), 2 ops in flight per lane, for the
//     id-embedding stream copy
//   * non-temporal (TH=NT) loads/stores: 1.18GB of single-touch streams must
//     not displace the 192MB L2
// Problem shape is fixed by the reference: B=8, P=25000, C=81, E=256.
// -----------------------------------------------------------------------------

typedef float v4f __attribute__((ext_vector_type(4)));

#define BBOX_CLIP 4.135166556742356f  // log(1000/16)

constexpr int kC = 81;      // classes (incl. background)
constexpr int kCM1 = 80;    // foreground classes
constexpr int kP = 25000;   // proposals per image
constexpr int kB = 8;       // batch
constexpr int kN = kB * kP; // 200000 rows
constexpr long long kE = 256;

__device__ __forceinline__ float wave_max32(float v) {
#pragma unroll
  for (int m = 16; m >= 1; m >>= 1)
    v = fmaxf(v, __shfl_xor(v, m, 32));
  return v;
}
__device__ __forceinline__ float wave_sum32(float v) {
#pragma unroll
  for (int m = 16; m >= 1; m >>= 1)
    v += __shfl_xor(v, m, 32);
  return v;
}

// Decode + clip + emit one (row, class) pair. Fully inlined; coalesced:
// consecutive lanes hit consecutive classes -> 512B bursts per wave on both
// the box_regression float4 load and the boxes float4 store.
__device__ __forceinline__ void emit_class(
    int c, float score, int p_idx, long long obase,
    const float* __restrict__ brow, float w, float h, float cx, float cy,
    float Wf, float Hf, float* __restrict__ boxes_out,
    float* __restrict__ scores_out, int* __restrict__ labels_out,
    int* __restrict__ rois_out) {
  const v4f d = __builtin_nontemporal_load((const v4f*)(brow + c * 4));
  const float dx = d.x * 0.1f;                    // /WX
  const float dy = d.y * 0.1f;                    // /WY
  const float dw = fminf(d.z * 0.2f, BBOX_CLIP);  // /WW, clipped
  const float dh = fminf(d.w * 0.2f, BBOX_CLIP);  // /WH, clipped
  const float pcx = fmaf(dx, w, cx);
  const float pcy = fmaf(dy, h, cy);
  const float pw = 0.5f * expf(dw) * w;
  const float ph = 0.5f * expf(dh) * h;
  v4f ob;
  ob.x = fminf(fmaxf(pcx - pw, 0.0f), Wf);
  ob.y = fminf(fmaxf(pcy - ph, 0.0f), Hf);
  ob.z = fminf(fmaxf(pcx + pw, 0.0f), Wf);
  ob.w = fminf(fmaxf(pcy + ph, 0.0f), Hf);
  const long long o = obase + (c - 1);
  __builtin_nontemporal_store(ob, (v4f*)(boxes_out + o * 4));
  __builtin_nontemporal_store(score, scores_out + o);
  __builtin_nontemporal_store(c, labels_out + o);
  __builtin_nontemporal_store(p_idx, rois_out + o);
}

// One wave32 per proposal row; 8 waves / 256-thread block; 25000 blocks.
// Lane covers classes {lane, lane+32, lane+64}: slots 0 and 1 are valid for
// every lane at C=81 (straight-line), slot 2 only for lane<17.
__global__ void __launch_bounds__(256)
rcnn_postproc_kernel(const float* __restrict__ logits,   // [N,81]
                     const float* __restrict__ boxreg,   // [N,324]
                     const float* __restrict__ props,    // [N,4]
                     const int* __restrict__ img_h_p,
                     const int* __restrict__ img_w_p,
                     float* __restrict__ boxes_out,      // [N,80,4]
                     float* __restrict__ scores_out,     // [N,80]
                     int* __restrict__ labels_out,       // [N,80]
                     int* __restrict__ rois_out) {       // [N,80]
  const int lane = threadIdx.x & 31;
  const int n = blockIdx.x * 8 + (threadIdx.x >> 5);  // exact: 25000*8 == kN

  const float Wf = (float)*img_w_p;
  const float Hf = (float)*img_h_p;

  // ---- softmax over 81 logits, 3 per lane (slot2 valid iff lane<17) ----
  const float* lrow = logits + (long long)n * kC;
  const bool has2 = lane < (kC - 64);
  const float l0 = __builtin_nontemporal_load(lrow + lane);
  const float l1 = __builtin_nontemporal_load(lrow + lane + 32);
  const float l2 = has2 ? __builtin_nontemporal_load(lrow + lane + 64)
                        : -3.0e38f;

  const float m = wave_max32(fmaxf(fmaxf(l0, l1), l2));
  const float e0 = expf(l0 - m);
  const float e1 = expf(l1 - m);
  const float e2 = has2 ? expf(l2 - m) : 0.0f;
  const float inv = 1.0f / wave_sum32(e0 + e1 + e2);

  // ---- proposal geometry (16B broadcast load) ----
  const v4f pr = *(const v4f*)(props + (long long)n * 4);
  const float w = pr.z - pr.x;
  const float h = pr.w - pr.y;
  const float cx = pr.x + 0.5f * w;
  const float cy = pr.y + 0.5f * h;

  const int p_idx = n % kP;  // constant divisor -> magic multiply
  const long long obase = (long long)n * kCM1;
  const float* brow = boxreg + (long long)n * (kC * 4);

  if (lane >= 1)  // slot 0: class = lane (class 0 = background, dropped)
    emit_class(lane, e0 * inv, p_idx, obase, brow, w, h, cx, cy, Wf, Hf,
               boxes_out, scores_out, labels_out, rois_out);
  // slot 1: class = lane+32, valid for all lanes -> straight-line
  emit_class(lane + 32, e1 * inv, p_idx, obase, brow, w, h, cx, cy, Wf, Hf,
             boxes_out, scores_out, labels_out, rois_out);
  if (has2)       // slot 2: class = lane+64, lanes 0..16
    emit_class(lane + 64, e2 * inv, p_idx, obase, brow, w, h, cx, cy, Wf, Hf,
               boxes_out, scores_out, labels_out, rois_out);
}

// id-embedding passthrough: 32B per lane streamed through LDS with the CDNA5
// async global<->LDS path. Two b128 async loads in flight (IOFFSET applies to
// both the LDS and global address, CDNA5 ISA), one
// s_wait_asynccnt, two async stores. HBM-bound; VGPR file untouched.
__global__ void __launch_bounds__(256)
id_copy_async_kernel(const float* __restrict__ src, float* __restrict__ dst,
                     long long nchunk /* # of 32B chunks */) {
  __shared__ char lds_buf[256 * 32];
  const long long i = (long long)blockIdx.x * 256 + threadIdx.x;
  if (i >= nchunk) return;

  const char* gsrc = (const char*)src + i * 32;
  char* gdst = (char*)dst + i * 32;
  const unsigned lds_addr =
      (unsigned)(uintptr_t)(&lds_buf[threadIdx.x * 32]);  // low 32b = LDS offset

  asm volatile("global_load_async_to_lds_b128 %0, %1, off"
               :
               : "v"(lds_addr), "v"(gsrc)
               : "memory");
  asm volatile("global_load_async_to_lds_b128 %0, %1, off offset:16"
               :
               : "v"(lds_addr), "v"(gsrc)
               : "memory");
  asm volatile("s_wait_asynccnt 0" ::: "memory");
  asm volatile("global_store_async_from_lds_b128 %0, %1, off"
               :
               : "v"(gdst), "v"(lds_addr)
               : "memory");
  asm volatile("global_store_async_from_lds_b128 %0, %1, off offset:16"
               :
               : "v"(gdst), "v"(lds_addr)
               : "memory");
  asm volatile("s_wait_asynccnt 0" ::: "memory");
}

extern "C" void kernel_launch(void* const* d_in, const int* in_sizes, int n_in,
                              void* d_out, int out_size, void* d_ws,
                              size_t ws_size, hipStream_t stream) {
  const float* class_logits = (const float*)d_in[0];  // [N,81]
  const float* box_reg      = (const float*)d_in[1];  // [N,324]
  const float* id_emb       = (const float*)d_in[2];  // [N,256]
  const float* proposals    = (const float*)d_in[3];  // [8,25000,4]
  const int*   img_h        = (const int*)d_in[4];
  const int*   img_w        = (const int*)d_in[5];

  float* out = (float*)d_out;
  const size_t boxes_elems = (size_t)kN * kCM1 * 4;  // 64,000,000
  const size_t sc_elems    = (size_t)kN * kCM1;      // 16,000,000
  float* boxes_out  = out;
  float* scores_out = out + boxes_elems;
  float* id_out     = scores_out + sc_elems;
  int*   labels_out = (int*)(id_out + (size_t)kN * kE);
  int*   rois_out   = labels_out + sc_elems;

  // Kernel A: one wave32 per row, 8 waves/block -> exactly 25000 blocks.
  rcnn_postproc_kernel<<<kN / 8, 256, 0, stream>>>(
      class_logits, box_reg, proposals, img_h, img_w, boxes_out, scores_out,
      labels_out, rois_out);

  // Kernel B: async-LDS stream copy of id embeddings (N*E*4 bytes, 32B/lane).
  const long long nchunk = (long long)kN * kE * 4 / 32;  // 6,400,000
  const long long gridB = (nchunk + 255) / 256;          // 25,000
  id_copy_async_kernel<<<(unsigned)gridB, 256, 0, stream>>>(id_emb, id_out,
                                                            nchunk);
}